// LanguageModelCriterion_binary_2_layer_22067541967244
// MI455X (gfx1250) — compile-verified
//
#include <hip/hip_runtime.h>
#include <cstdint>

// ---------------------------------------------------------------------------
// 2-level hierarchical softmax loss (adaptive-softmax style), MI455X/gfx1250.
// Memory/latency-bound (~6.5 MB of logit reads for 8192 tokens):
//   - one wave32 per token
//   - DOUBLE-BUFFERED async global->LDS staging of the two 99-float slices
//     (ASYNCcnt completes in-order => s_wait_asynccnt 4 overlaps next token's
//      DMA with current token's exp/log compute)
//   - wave shuffles for the 100-wide logsumexp
//   - deterministic two-kernel reduction (no atomics)
// ---------------------------------------------------------------------------

#define WAVES_PER_BLOCK 8
#define BLOCK_THREADS   (WAVES_PER_BLOCK * 32)
#define GRID_BLOCKS     128
#define SLICE_PAD       128          // per-slice LDS floats (>= 99, 16B aligned)
#define NEG_INF_F       (-3.402823466e38f)

// ---- wave32 reductions (lane-width 32, never 64) --------------------------
__device__ __forceinline__ float wave_max_f32(float v) {
#pragma unroll
  for (int off = 16; off > 0; off >>= 1)
    v = fmaxf(v, __shfl_xor(v, off, 32));
  return v;
}

__device__ __forceinline__ float wave_sum_f32(float v) {
#pragma unroll
  for (int off = 16; off > 0; off >>= 1)
    v += __shfl_xor(v, off, 32);
  return v;
}

// ---- async global -> LDS copy of one contiguous f32 slice -----------------
// nvalid <= 128. Bulk: (nvalid/4) lanes move 16B each via async B128;
// tail: (nvalid%4) lanes move 4B each via async B32. ASYNCcnt-tracked.
// Issues exactly (nvalid>=4) + (nvalid%4!=0) async instructions per wave.
__device__ __forceinline__ void async_slice(const float* __restrict__ g,
                                            uint32_t lds_byte, int nvalid,
                                            int lane) {
  const int nb128 = nvalid >> 2;
  if (lane < nb128) {
    uint64_t ga = (uint64_t)(g + lane * 4);
    uint32_t la = lds_byte + (uint32_t)lane * 16u;
    asm volatile("global_load_async_to_lds_b128 %0, %1, off"
                 :: "v"(la), "v"(ga) : "memory");
  } else if (lane < nb128 + (nvalid & 3)) {
    const int j = (nb128 << 2) + (lane - nb128);
    uint64_t ga = (uint64_t)(g + j);
    uint32_t la = lds_byte + (uint32_t)j * 4u;
    asm volatile("global_load_async_to_lds_b32 %0, %1, off"
                 :: "v"(la), "v"(ga) : "memory");
  }
}

// ---- log_softmax([slice | 0])[take] for one wave, slice resident in LDS ---
__device__ __forceinline__ float lse_taken(const float* __restrict__ s,
                                           int nvalid, int take, int lane) {
  float x[4];
  float m = 0.0f;                      // implicit appended zero logit
#pragma unroll
  for (int k = 0; k < 4; ++k) {
    const int i = lane + 32 * k;
    x[k] = (i < nvalid) ? s[i] : NEG_INF_F;
    m = fmaxf(m, x[k]);
  }
  m = wave_max_f32(m);                 // m >= 0 always
  float e = 0.0f;
#pragma unroll
  for (int k = 0; k < 4; ++k) {
    const int i = lane + 32 * k;
    e += (i < nvalid) ? __expf(x[k] - m) : 0.0f;
  }
  const float ssum = wave_sum_f32(e) + __expf(0.0f - m);  // + implicit-zero term
  const float val = (take < nvalid) ? s[take] : 0.0f;     // take==nvalid -> 0
  return (val - m) - __logf(ssum);
}

// ---- kernel 1: per-block partial (loss, mask) sums ------------------------
__global__ __launch_bounds__(BLOCK_THREADS)
void hsm_loss_partial_kernel(const float* __restrict__ input,      // [N, V1]
                             const int*   __restrict__ target,     // [N]
                             const float* __restrict__ mask,       // [N]
                             const int*   __restrict__ vocab2code, // [V, 2]
                             float* __restrict__ partial,          // [2*grid]
                             int N, int V1, int nc, int cs) {
  __shared__ float sbuf[WAVES_PER_BLOCK][2][2][SLICE_PAD];  // [wave][pingpong][level]
  __shared__ float sred[WAVES_PER_BLOCK][2];

  const int lane = threadIdx.x & 31;
  const int w    = threadIdx.x >> 5;
  const int waveGlobal = blockIdx.x * WAVES_PER_BLOCK + w;
  const int totalWaves = gridDim.x * WAVES_PER_BLOCK;

  const int n1 = nc - 1;                // level-1 real logits (99)
  const int n2 = cs - 1;                // level-2 real logits (99)

  // async instructions issued per token (4 when both slices need bulk+tail)
  const int perTok = ((n1 >= 4) ? 1 : 0) + ((n1 & 3) ? 1 : 0) +
                     ((n2 >= 4) ? 1 : 0) + ((n2 & 3) ? 1 : 0);

  float* sp[2][2];
  uint32_t lb[2][2];
#pragma unroll
  for (int b = 0; b < 2; ++b)
#pragma unroll
    for (int lv = 0; lv < 2; ++lv) {
      sp[b][lv] = &sbuf[w][b][lv][0];
      lb[b][lv] = (uint32_t)(uintptr_t)sp[b][lv];  // wave-relative LDS bytes
    }

  float lossAcc = 0.0f, maskAcc = 0.0f;

  int t = waveGlobal;            // wave-uniform
  int cur = 0;
  int c0 = 0, c2 = 0;
  float mk = 0.0f;

  if (t < N) {                   // prologue: stage token t into buffer 0
    const int tgt = target[t];
    c0 = vocab2code[2 * tgt + 0];
    c2 = vocab2code[2 * tgt + 1];
    mk = mask[t];
    const float* row = input + (size_t)t * (size_t)V1;
    async_slice(row, lb[cur][0], n1, lane);
    async_slice(row + n1 + (size_t)c0 * (size_t)n2, lb[cur][1], n2, lane);
  }

  while (t < N) {
    const int  tn      = t + totalWaves;
    const bool hasNext = (tn < N);
    int c0n = 0, c2n = 0;
    float mkn = 0.0f;

    if (hasNext) {
      const int tgtn = target[tn];
      c0n = vocab2code[2 * tgtn + 0];
      c2n = vocab2code[2 * tgtn + 1];
      mkn = mask[tn];
      // Prior ds reads of buffer cur^1 must land before the DMA engine
      // overwrites it (no HW ordering between wave LDS reads and async writes).
      asm volatile("s_wait_dscnt 0x0" ::: "memory");
      const float* rown = input + (size_t)tn * (size_t)V1;
      async_slice(rown, lb[cur ^ 1][0], n1, lane);
      async_slice(rown + n1 + (size_t)c0n * (size_t)n2, lb[cur ^ 1][1], n2, lane);

      // warm L2 two tokens ahead while both DMA groups are in flight
      const int tp = tn + totalWaves;
      if (tp < N && lane < 4)
        __builtin_prefetch(input + (size_t)tp * (size_t)V1 + lane * 32, 0, 0);
    }

    // In-order ASYNCcnt: leaving <= perTok outstanding guarantees the OLDER
    // group (token t) has fully landed while token tn's copies keep flying.
    if (hasNext && perTok == 4)
      asm volatile("s_wait_asynccnt 0x4" ::: "memory");
    else
      asm volatile("s_wait_asynccnt 0x0" ::: "memory");

    const float l1 = lse_taken(sp[cur][0], n1, c0, lane);
    const float l2 = lse_taken(sp[cur][1], n2, c2, lane);
    lossAcc += (l1 + l2) * mk;
    maskAcc += mk;

    t = tn; c0 = c0n; c2 = c2n; mk = mkn;
    cur ^= 1;
  }

  // lane-uniform after the shuffles: lane 0 publishes the wave result
  if (lane == 0) { sred[w][0] = lossAcc; sred[w][1] = maskAcc; }
  __syncthreads();
  if (threadIdx.x == 0) {
    float L = 0.0f, M = 0.0f;
#pragma unroll
    for (int i = 0; i < WAVES_PER_BLOCK; ++i) { L += sred[i][0]; M += sred[i][1]; }
    partial[2 * blockIdx.x + 0] = -L;   // loss sign applied here
    partial[2 * blockIdx.x + 1] = M;
  }
}

// ---- kernel 2: deterministic fixed-order fold + normalize -----------------
__global__ void hsm_finalize_kernel(const float* __restrict__ partial,
                                    int nblocks, float* __restrict__ out) {
  if (blockIdx.x == 0 && threadIdx.x == 0) {
    float L = 0.0f, M = 0.0f;
    for (int i = 0; i < nblocks; ++i) {
      L += partial[2 * i + 0];
      M += partial[2 * i + 1];
    }
    out[0] = L / M;
  }
}

// ---------------------------------------------------------------------------
extern "C" void kernel_launch(void* const* d_in, const int* in_sizes, int n_in,
                              void* d_out, int out_size, void* d_ws, size_t ws_size,
                              hipStream_t stream) {
  const float* input      = (const float*)d_in[0];  // [B, L, VOCAB-1] f32
  const int*   target     = (const int*)  d_in[1];  // [B, L] i32
  const float* mask       = (const float*)d_in[2];  // [B, L] f32
  const int*   vocab2code = (const int*)  d_in[3];  // [VOCAB, 2] i32
  // d_in[4] = cluster_size (uniform; only its count is needed)
  // d_in[5] = depth (== 2)

  const int N     = in_sizes[1];             // B*L tokens
  const int V1    = in_sizes[0] / N;         // VOCAB-1 logits per token
  const int nc    = in_sizes[4];             // N_CLUSTER
  const int vocab = in_sizes[3] / 2;         // VOCAB
  const int cs    = vocab / nc;              // uniform cluster size

  float* partial = (float*)d_ws;             // 2*GRID_BLOCKS floats of scratch

  hipLaunchKernelGGL(hsm_loss_partial_kernel,
                     dim3(GRID_BLOCKS), dim3(BLOCK_THREADS), 0, stream,
                     input, target, mask, vocab2code, partial, N, V1, nc, cs);

  hipLaunchKernelGGL(hsm_finalize_kernel, dim3(1), dim3(32), 0, stream,
                     partial, GRID_BLOCKS, (float*)d_out);

  (void)n_in; (void)out_size; (void)ws_size;
}